// NearestColorDistanceLoss_12704513261609
// MI455X (gfx1250) — compile-verified
//
#include <hip/hip_runtime.h>
#include <hip/hip_bf16.h>

typedef float v2f __attribute__((ext_vector_type(2)));
typedef float v8f __attribute__((ext_vector_type(8)));

// Zero the scalar output before accumulation (harness poisons d_out).
__global__ void ncdl_zero_out(float* out) {
    if (threadIdx.x == 0) out[0] = 0.0f;
}

// Each wave (32 lanes) processes a 16-row tile of output colors.
// Block = 256 threads = 8 waves = 128 colors per block.
// Palette: M == 128 (8 column tiles of 16), preprocessed into LDS as
// records (-2p0, -2p1, -2p2, |p|^2).  dist^2(m,n) = |o_m|^2 + D[m][n]
// where D = A x B with A row = (o0,o1,o2,1), B col = (-2p0,-2p1,-2p2,|p|^2).
__global__ __launch_bounds__(256)
void ncdl_wmma_kernel(const float* __restrict__ colors,
                      const float* __restrict__ palette,
                      float* __restrict__ out,
                      float inv_n) {
    __shared__ float lpal[128 * 4];

    // --- build palette records in LDS (128 entries, thread t -> entry t) ---
    int t = threadIdx.x;
    if (t < 128) {
        float p0 = palette[3 * t + 0];
        float p1 = palette[3 * t + 1];
        float p2 = palette[3 * t + 2];
        lpal[4 * t + 0] = -2.0f * p0;
        lpal[4 * t + 1] = -2.0f * p1;
        lpal[4 * t + 2] = -2.0f * p2;
        lpal[4 * t + 3] = p0 * p0 + p1 * p1 + p2 * p2;
    }
    __syncthreads();

    const int lane    = threadIdx.x & 31;
    const int waveId  = threadIdx.x >> 5;
    const int rowBase = blockIdx.x * 128 + waveId * 16;
    const int subRow  = lane & 15;       // row/col index this lane feeds
    const int row     = rowBase + subRow;
    const bool hiHalf = lane >= 16;

    // Prefetch next block's slice of the color stream into cache.
    __builtin_prefetch(colors + 3 * (row + 128), 0, 1);

    // --- A operand: 16x4 fp32, row = (o0,o1,o2,1) ---
    // lanes 0-15: (K0,K1) = (o0,o1); lanes 16-31: (K2,K3) = (o2,1)
    const float* cp = colors + 3 * row;
    v2f a;
    if (!hiHalf) { a.x = cp[0]; a.y = cp[1]; }
    else         { a.x = cp[2]; a.y = 1.0f;  }

    // |o_m|^2: lane L holds R^2+G^2 (low half) or B^2 (high half) for row L&15
    float partial = hiHalf ? (a.x * a.x) : (a.x * a.x + a.y * a.y);
    float norm2_all = partial + __shfl_xor(partial, 16, 32);  // row (lane&15) norm^2

    // --- 8 palette tiles of 16 columns: straight-line WMMA + elementwise min ---
    v8f minE;
    #pragma unroll
    for (int tt = 0; tt < 8; ++tt) {
        const int base = (tt * 16 + subRow) * 4;
        v2f b;
        if (!hiHalf) { b.x = lpal[base + 0]; b.y = lpal[base + 1]; }
        else         { b.x = lpal[base + 2]; b.y = lpal[base + 3]; }

        v8f c = {};
        v8f d = __builtin_amdgcn_wmma_f32_16x16x4_f32(
            /*neg_a=*/false, a, /*neg_b=*/false, b,
            /*c_mod=*/(short)0, c, /*reuse_a=*/false, /*reuse_b=*/false);

        if (tt == 0) {
            minE = d;
        } else {
            #pragma unroll
            for (int i = 0; i < 8; ++i) minE[i] = __builtin_fminf(minE[i], d[i]);
        }
    }

    // --- min across the 16 columns (each 16-lane half holds N=0..15) ---
    #pragma unroll
    for (int mask = 1; mask <= 8; mask <<= 1) {
        #pragma unroll
        for (int i = 0; i < 8; ++i) {
            float o = __shfl_xor(minE[i], mask, 32);
            minE[i] = __builtin_fminf(minE[i], o);
        }
    }
    // Now: lanes 0-15 hold row-mins for rows rowBase+0..7 (vgpr i -> row i),
    //      lanes 16-31 hold rows rowBase+8..15.

    float me[8];
    #pragma unroll
    for (int i = 0; i < 8; ++i) me[i] = minE[i];

    const bool leader  = (lane & 15) == 0;   // lanes 0 and 16 accumulate
    const int halfBase = (lane >> 4) * 8;    // 0 or 8
    float wsum = 0.0f;
    #pragma unroll
    for (int r = 0; r < 8; ++r) {
        float n2   = __shfl(norm2_all, halfBase + r, 32); // |o|^2 of row halfBase+r
        float d2   = me[r] + n2;
        float dist = __builtin_sqrtf(__builtin_fmaxf(d2, 0.0f));
        if (leader) wsum += dist;
    }

    // combine lane0 + lane16, one atomic per wave
    wsum += __shfl_xor(wsum, 16, 32);
    if (lane == 0) atomicAdd(out, wsum * inv_n);
}

extern "C" void kernel_launch(void* const* d_in, const int* in_sizes, int n_in,
                              void* d_out, int out_size, void* d_ws, size_t ws_size,
                              hipStream_t stream) {
    const float* colors  = (const float*)d_in[0];   // (N,3) fp32
    const float* palette = (const float*)d_in[1];   // (M,3) fp32, M == 128
    float* out = (float*)d_out;                     // scalar fp32

    const int n = in_sizes[0] / 3;   // 131072

    ncdl_zero_out<<<1, 32, 0, stream>>>(out);

    const int rowsPerBlock = 128;                  // 8 waves * 16 rows
    const int blocks = n / rowsPerBlock;           // N is an exact multiple
    ncdl_wmma_kernel<<<blocks, 256, 0, stream>>>(colors, palette, out,
                                                 1.0f / (float)n);
    (void)d_ws; (void)ws_size; (void)n_in; (void)out_size; (void)in_sizes;
}